// CrossModalGatedAttention_27882927686024
// MI455X (gfx1250) — compile-verified
//
#include <hip/hip_runtime.h>
#include <hip/hip_bf16.h>
#include <math.h>

typedef float v2f __attribute__((ext_vector_type(2)));
typedef float v8f __attribute__((ext_vector_type(8)));

#define BB 128
#define TT 1024
#define DD 1024
#define NCH 4
#define TCHUNK (TT / NCH)   // 256 t per block in the streaming pass

// ---------------------------------------------------------------------------
// WMMA fp32 GEMM: C[M,N] = A[M,K] @ op(B)[K,N] (+ fused epilogue)
// block = 256 threads (8 waves); wave w -> 16x16 tile at cols blockIdx.x*128+w*16,
// rows blockIdx.y*16. Uses V_WMMA_F32_16X16X4_F32 (fp32 in, fp32 acc).
// Fragment mapping (ISA 7.12.2, 32-bit): for A(16x4) and B(4x16),
//   k = k0 + 2*(lane>>4) + vgpr ;  m/n = lane&15
// C/D(16x16): element (m = vgpr + 8*(lane>>4), n = lane&15)
// EPI: 0 = none, 1 = +bias1, 2 = sigmoid(x + bias1 + bias2)
// ---------------------------------------------------------------------------
template <int TRANSB, int EPI>
__global__ __launch_bounds__(256)
void wmma_gemm_f32(const float* __restrict__ A, const float* __restrict__ B,
                   float* __restrict__ C, int K, int lda, int ldb, int ldc,
                   const float* __restrict__ bias1, const float* __restrict__ bias2)
{
  const int wave = threadIdx.x >> 5;
  const int lane = threadIdx.x & 31;
  const int ln   = lane & 15;       // n (for B/C) or m-row (for A)
  const int g    = lane >> 4;       // lane group
  const int row0 = blockIdx.y << 4;
  const int col0 = (blockIdx.x << 7) + (wave << 4);

  const float* Arow = A + (size_t)(row0 + ln) * lda;
  v8f acc = {};

  #pragma unroll 4
  for (int k0 = 0; k0 < K; k0 += 4) {
    const int kb = k0 + (g << 1);
    v2f a;
    a.x = Arow[kb];
    a.y = Arow[kb + 1];
    v2f b;
    if (TRANSB) {
      const float* Bp = B + (size_t)(col0 + ln) * ldb + kb;
      b.x = Bp[0];
      b.y = Bp[1];
    } else {
      const float* Bp = B + (size_t)kb * ldb + col0 + ln;
      b.x = Bp[0];
      b.y = Bp[ldb];
    }
    acc = __builtin_amdgcn_wmma_f32_16x16x4_f32(
        /*neg_a=*/false, a, /*neg_b=*/false, b,
        /*c_mod=*/(short)0, acc, /*reuse_a=*/false, /*reuse_b=*/false);
  }

  const int n = col0 + ln;
  float badd = 0.0f;
  if (EPI >= 1) badd += bias1[n];
  if (EPI == 2) badd += bias2[n];

  #pragma unroll
  for (int j = 0; j < 8; ++j) {
    const int m = row0 + j + (g << 3);
    float v = acc[j] + badd;
    if (EPI == 2) v = 1.0f / (1.0f + __expf(-v));
    C[(size_t)m * ldc + n] = v;
  }
}

// ---------------------------------------------------------------------------
// Fused scores + online softmax + attention pooling (single pass over z_eeg).
// grid = (B, NCH); block = 256 (8 waves). Wave wi handles t = chunk_base + wi + 8*i.
// Each wave keeps running (max m, sum l, acc[1024] as 8x float4/lane);
// block combine across 8 waves through LDS; per-chunk partial to workspace.
// scores scale 1/sqrt(1024)=1/32 folded into the Qk register preload.
// ---------------------------------------------------------------------------
__global__ __launch_bounds__(256)
void attn_pool_partial(const float* __restrict__ z_eeg, const float* __restrict__ Qk,
                       float* __restrict__ part_acc, float* __restrict__ part_ml)
{
  const int b    = blockIdx.x;
  const int ch   = blockIdx.y;
  const int wave = threadIdx.x >> 5;
  const int lane = threadIdx.x & 31;

  __shared__ float s_acc[8][DD];   // 32 KB
  __shared__ float s_m[8];
  __shared__ float s_l[8];

  const float invs = 1.0f / 32.0f; // 1/sqrt(D)
  float4 qk[8];
  const float4* Qk4 = (const float4*)(Qk + (size_t)b * DD);
  #pragma unroll
  for (int j = 0; j < 8; ++j) {
    float4 q = Qk4[lane + 32 * j];
    qk[j].x = q.x * invs; qk[j].y = q.y * invs;
    qk[j].z = q.z * invs; qk[j].w = q.w * invs;
  }

  float mrun = -3.402823466e38f;
  float lrun = 0.0f;
  float4 acc[8];
  #pragma unroll
  for (int j = 0; j < 8; ++j) { acc[j].x = acc[j].y = acc[j].z = acc[j].w = 0.0f; }

  const int tbase = ch * TCHUNK + wave;
  for (int i = 0; i < TCHUNK / 8; ++i) {      // 32 rows per wave
    const int t = tbase + (i << 3);
    const float4* Z4 = (const float4*)(z_eeg + ((size_t)b * TT + t) * DD);
    float4 zv[8];
    #pragma unroll
    for (int j = 0; j < 8; ++j) zv[j] = Z4[lane + 32 * j];

    float s = 0.0f;
    #pragma unroll
    for (int j = 0; j < 8; ++j) {
      s = fmaf(zv[j].x, qk[j].x, s);
      s = fmaf(zv[j].y, qk[j].y, s);
      s = fmaf(zv[j].z, qk[j].z, s);
      s = fmaf(zv[j].w, qk[j].w, s);
    }
    #pragma unroll
    for (int off = 16; off > 0; off >>= 1)
      s += __shfl_xor(s, off, 32);            // wave32 allreduce

    const float mn = fmaxf(mrun, s);
    const float sc = __expf(mrun - mn);
    const float p  = __expf(s - mn);
    lrun = fmaf(lrun, sc, p);
    #pragma unroll
    for (int j = 0; j < 8; ++j) {
      acc[j].x = fmaf(acc[j].x, sc, p * zv[j].x);
      acc[j].y = fmaf(acc[j].y, sc, p * zv[j].y);
      acc[j].z = fmaf(acc[j].z, sc, p * zv[j].z);
      acc[j].w = fmaf(acc[j].w, sc, p * zv[j].w);
    }
    mrun = mn;
  }

  if (lane == 0) { s_m[wave] = mrun; s_l[wave] = lrun; }
  float4* sa = (float4*)&s_acc[wave][0];
  #pragma unroll
  for (int j = 0; j < 8; ++j) sa[lane + 32 * j] = acc[j];
  __syncthreads();

  float M = -3.402823466e38f;
  #pragma unroll
  for (int w = 0; w < 8; ++w) M = fmaxf(M, s_m[w]);
  float L = 0.0f;
  float fw[8];
  #pragma unroll
  for (int w = 0; w < 8; ++w) {
    fw[w] = __expf(s_m[w] - M);
    L = fmaf(s_l[w], fw[w], L);
  }

  const int tid = threadIdx.x;                // 256 threads * float4 = 1024 dims
  float4 sum; sum.x = sum.y = sum.z = sum.w = 0.0f;
  #pragma unroll
  for (int w = 0; w < 8; ++w) {
    float4 v = ((float4*)&s_acc[w][0])[tid];
    sum.x = fmaf(v.x, fw[w], sum.x);
    sum.y = fmaf(v.y, fw[w], sum.y);
    sum.z = fmaf(v.z, fw[w], sum.z);
    sum.w = fmaf(v.w, fw[w], sum.w);
  }
  ((float4*)(part_acc + (size_t)(ch * BB + b) * DD))[tid] = sum;
  if (tid == 0) {
    part_ml[(ch * BB + b) * 2 + 0] = M;
    part_ml[(ch * BB + b) * 2 + 1] = L;
  }
}

// Combine NCH chunk partials -> A[b, :]
__global__ __launch_bounds__(256)
void attn_pool_combine(const float* __restrict__ part_acc,
                       const float* __restrict__ part_ml,
                       float* __restrict__ Aout)
{
  const int b   = blockIdx.x;
  const int tid = threadIdx.x;
  float M = -3.402823466e38f;
  #pragma unroll
  for (int ch = 0; ch < NCH; ++ch) M = fmaxf(M, part_ml[(ch * BB + b) * 2]);
  float L = 0.0f;
  float fc[NCH];
  #pragma unroll
  for (int ch = 0; ch < NCH; ++ch) {
    fc[ch] = __expf(part_ml[(ch * BB + b) * 2] - M);
    L = fmaf(part_ml[(ch * BB + b) * 2 + 1], fc[ch], L);
  }
  const float invL = 1.0f / L;
  float4 s; s.x = s.y = s.z = s.w = 0.0f;
  #pragma unroll
  for (int ch = 0; ch < NCH; ++ch) {
    float4 v = ((const float4*)(part_acc + (size_t)(ch * BB + b) * DD))[tid];
    s.x = fmaf(v.x, fc[ch], s.x);
    s.y = fmaf(v.y, fc[ch], s.y);
    s.z = fmaf(v.z, fc[ch], s.z);
    s.w = fmaf(v.w, fc[ch], s.w);
  }
  float4 o; o.x = s.x * invL; o.y = s.y * invL; o.z = s.z * invL; o.w = s.w * invL;
  ((float4*)(Aout + (size_t)b * DD))[tid] = o;
}

// concat = [A, z_rppg] per row (materialized so the Wf GEMM is one K=2048 pass)
__global__ __launch_bounds__(256)
void concat_kernel(const float* __restrict__ Ap, const float* __restrict__ zr,
                   float* __restrict__ cc)
{
  const int b = blockIdx.x, tid = threadIdx.x;
  ((float4*)(cc + (size_t)b * 2 * DD))[tid]      = ((const float4*)(Ap + (size_t)b * DD))[tid];
  ((float4*)(cc + (size_t)b * 2 * DD + DD))[tid] = ((const float4*)(zr + (size_t)b * DD))[tid];
}

// h = relu(z_rppg + m * f)   (m has Wm_b fused; f is post-sigmoid)
__global__ __launch_bounds__(256)
void fuse_kernel(const float* __restrict__ zr, const float* __restrict__ m,
                 const float* __restrict__ f, float* __restrict__ h)
{
  const int b = blockIdx.x, tid = threadIdx.x;
  float4 z = ((const float4*)(zr + (size_t)b * DD))[tid];
  float4 mv = ((const float4*)(m + (size_t)b * DD))[tid];
  float4 fv = ((const float4*)(f + (size_t)b * DD))[tid];
  float4 o;
  o.x = fmaxf(fmaf(mv.x, fv.x, z.x), 0.0f);
  o.y = fmaxf(fmaf(mv.y, fv.y, z.y), 0.0f);
  o.z = fmaxf(fmaf(mv.z, fv.z, z.z), 0.0f);
  o.w = fmaxf(fmaf(mv.w, fv.w, z.w), 0.0f);
  ((float4*)(h + (size_t)b * DD))[tid] = o;
}

extern "C" void kernel_launch(void* const* d_in, const int* in_sizes, int n_in,
                              void* d_out, int out_size, void* d_ws, size_t ws_size,
                              hipStream_t stream) {
  const float* z_eeg  = (const float*)d_in[0];
  const float* z_rppg = (const float*)d_in[1];
  const float* Wq     = (const float*)d_in[2];
  const float* Wk     = (const float*)d_in[3];
  const float* Wm_w   = (const float*)d_in[4];
  const float* Wm_b   = (const float*)d_in[5];
  const float* Wf_w   = (const float*)d_in[6];
  const float* Wf_b   = (const float*)d_in[7];
  const float* bf     = (const float*)d_in[8];
  float* out = (float*)d_out;
  float* ws  = (float*)d_ws;

  float* Q      = ws;                  // [128,1024]
  float* Qk     = Q + BB * DD;         // [128,1024]  (= Q @ Wk^T, absorbs Wk)
  float* Apool  = Qk + BB * DD;        // [128,1024]
  float* ccat   = Apool + BB * DD;     // [128,2048]
  float* Mv     = ccat + BB * 2 * DD;  // [128,1024]  m (+Wm_b)
  float* Fv     = Mv + BB * DD;        // [128,1024]  f (sigmoid)
  float* pacc   = Fv + BB * DD;        // [NCH,128,1024]
  float* pml    = pacc + NCH * BB * DD;// [NCH,128,2]

  const dim3 blk(256);
  const dim3 ggrid(DD / 128, BB / 16); // (8, 8)

  // Q = z_rppg @ Wq
  wmma_gemm_f32<0, 0><<<ggrid, blk, 0, stream>>>(z_rppg, Wq, Q, DD, DD, DD, DD, nullptr, nullptr);
  // Qk = Q @ Wk^T  (folds the K-projection into the query side)
  wmma_gemm_f32<1, 0><<<ggrid, blk, 0, stream>>>(Q, Wk, Qk, DD, DD, DD, DD, nullptr, nullptr);
  // single-pass online-softmax attention pooling over z_eeg
  attn_pool_partial<<<dim3(BB, NCH), blk, 0, stream>>>(z_eeg, Qk, pacc, pml);
  attn_pool_combine<<<dim3(BB), blk, 0, stream>>>(pacc, pml, Apool);
  // gate + modulation
  concat_kernel<<<dim3(BB), blk, 0, stream>>>(Apool, z_rppg, ccat);
  wmma_gemm_f32<0, 1><<<ggrid, blk, 0, stream>>>(Apool, Wm_w, Mv, DD, DD, DD, DD, Wm_b, nullptr);
  wmma_gemm_f32<0, 2><<<ggrid, blk, 0, stream>>>(ccat, Wf_w, Fv, 2 * DD, 2 * DD, DD, DD, Wf_b, bf);
  fuse_kernel<<<dim3(BB), blk, 0, stream>>>(z_rppg, Mv, Fv, out);
}